// GraphAttention_7275674600228
// MI455X (gfx1250) — compile-verified
//
#include <hip/hip_runtime.h>

typedef __attribute__((ext_vector_type(2))) float v2f;
typedef __attribute__((ext_vector_type(8))) float v8f;

#define B_    32
#define A_    1024
#define NN_   5
#define FIN_  128
#define E0_   128
#define E1_   32
#define H_    32
#define NOUT_ 32
#define NACT_ 8

__device__ __forceinline__ v8f vzero8() {
  v8f z = {0.f, 0.f, 0.f, 0.f, 0.f, 0.f, 0.f, 0.f};
  return z;
}

__device__ __forceinline__ v8f wmma4(v2f a, v2f b, v8f c) {
  // V_WMMA_F32_16X16X4_F32 : D = A(16x4) x B(4x16) + C(16x16)
  return __builtin_amdgcn_wmma_f32_16x16x4_f32(false, a, false, b, (short)0, c,
                                               false, false);
}

// ---------------------------------------------------------------------------
// Kernel 1: s[b,j,n] = sum_m y[b,j,n,m]   (pure HBM bandwidth, 671 MB read)
// One wave32 per row of 1024 floats; 8x float4 per lane, shuffle reduce.
// ---------------------------------------------------------------------------
__global__ __launch_bounds__(256) void ysum_kernel(const float* __restrict__ y,
                                                   float* __restrict__ s) {
  const int lane = threadIdx.x & 31;
  const int row  = (int)((blockIdx.x * blockDim.x + threadIdx.x) >> 5);
  const float4* p = (const float4*)(y + (size_t)row * A_);
  float acc = 0.f;
#pragma unroll
  for (int k = 0; k < (A_ / 4 / 32); ++k) {
    float4 v = p[lane + k * 32];
    acc += (v.x + v.y) + (v.z + v.w);
  }
#pragma unroll
  for (int off = 16; off > 0; off >>= 1)
    acc += __shfl_xor(acc, off, 32);
  if (lane == 0) s[row] = acc;
}

// ---------------------------------------------------------------------------
// Kernel 2: fused MLP + attention chain. 4 waves/block, 16 agents per wave.
// ---------------------------------------------------------------------------
__global__ __launch_bounds__(128) void gat_kernel(
    const float* __restrict__ x, const float* __restrict__ sbuf,
    const float* __restrict__ We0, const float* __restrict__ be0,
    const float* __restrict__ We1, const float* __restrict__ be1,
    const float* __restrict__ W1, const float* __restrict__ b1,
    const float* __restrict__ W2, const float* __restrict__ b2,
    const float* __restrict__ W3, const float* __restrict__ b3,
    const float* __restrict__ W4, const float* __restrict__ b4,
    const float* __restrict__ Wp, const float* __restrict__ bp,
    float* __restrict__ logits, float* __restrict__ att) {

  // h tile (E0=128 cols, padded to 132 to break bank conflicts),
  // later reused as per-agent vectors: A1[0..31] G2[32..63] G3[64..95] ctx[96..127]
  __shared__ float lds_h[4][16][132];
  // h2 tile (E1=32, padded), later reused for the `out` tile (NOUT=32)
  __shared__ float lds_h2[4][16][36];

  const int lane = threadIdx.x & 31;
  const int wid  = threadIdx.x >> 5;   // wave in block, 0..3
  const int half = lane >> 4;          // K-half selector for A/B fragments
  const int ln   = lane & 15;          // M (for A) / N (for B,C,D)
  const int g0   = blockIdx.x * 64 + wid * 16;  // first flat agent row (b*A + j)

  // ---------------- GEMM1: h = relu(x @ We0 + be0), [16x128], K=128 ---------
  const float* xrow = x + (size_t)(g0 + ln) * FIN_;
  v8f acc[8];
#pragma unroll
  for (int t = 0; t < 8; ++t) acc[t] = vzero8();
  for (int k0 = 0; k0 < FIN_; k0 += 4) {
    const int kr = k0 + 2 * half;
    v2f a; a.x = xrow[kr]; a.y = xrow[kr + 1];
#pragma unroll
    for (int t = 0; t < 8; ++t) {
      v2f bb;
      bb.x = We0[(size_t)kr * E0_ + t * 16 + ln];
      bb.y = We0[(size_t)(kr + 1) * E0_ + t * 16 + ln];
      acc[t] = wmma4(a, bb, acc[t]);
    }
  }
#pragma unroll
  for (int t = 0; t < 8; ++t) {
    float bv = be0[t * 16 + ln];
#pragma unroll
    for (int r = 0; r < 8; ++r) {
      int m = r + 8 * half;
      lds_h[wid][m][t * 16 + ln] = fmaxf(acc[t][r] + bv, 0.f);
    }
  }
  __syncthreads();

  // ---------------- GEMM2: h2 = relu(h @ We1 + be1), [16x32], K=128 ---------
  const float* hrow = &lds_h[wid][ln][0];
  v8f a2[2];
  a2[0] = vzero8(); a2[1] = vzero8();
  for (int k0 = 0; k0 < E0_; k0 += 4) {
    const int kr = k0 + 2 * half;
    v2f a; a.x = hrow[kr]; a.y = hrow[kr + 1];
#pragma unroll
    for (int t = 0; t < 2; ++t) {
      v2f bb;
      bb.x = We1[(size_t)kr * E1_ + t * 16 + ln];
      bb.y = We1[(size_t)(kr + 1) * E1_ + t * 16 + ln];
      a2[t] = wmma4(a, bb, a2[t]);
    }
  }
#pragma unroll
  for (int t = 0; t < 2; ++t) {
    float bv = be1[t * 16 + ln];
#pragma unroll
    for (int r = 0; r < 8; ++r) {
      int m = r + 8 * half;
      lds_h2[wid][m][t * 16 + ln] = fmaxf(a2[t][r] + bv, 0.f);
    }
  }
  __syncthreads();

  // ------------- A1 = relu(h2@W1+b1), G2 = h2@W2, G3 = h2@W3, K=32 ----------
  const float* h2row = &lds_h2[wid][ln][0];
  v8f aA[2], aB[2], aC[2];
#pragma unroll
  for (int t = 0; t < 2; ++t) { aA[t] = vzero8(); aB[t] = vzero8(); aC[t] = vzero8(); }
  for (int k0 = 0; k0 < E1_; k0 += 4) {
    const int kr = k0 + 2 * half;
    v2f a; a.x = h2row[kr]; a.y = h2row[kr + 1];
#pragma unroll
    for (int t = 0; t < 2; ++t) {
      v2f w;
      w.x = W1[(size_t)kr * H_ + t * 16 + ln];
      w.y = W1[(size_t)(kr + 1) * H_ + t * 16 + ln];
      aA[t] = wmma4(a, w, aA[t]);
      w.x = W2[(size_t)kr * H_ + t * 16 + ln];
      w.y = W2[(size_t)(kr + 1) * H_ + t * 16 + ln];
      aB[t] = wmma4(a, w, aB[t]);
      w.x = W3[(size_t)kr * H_ + t * 16 + ln];
      w.y = W3[(size_t)(kr + 1) * H_ + t * 16 + ln];
      aC[t] = wmma4(a, w, aC[t]);
    }
  }
  // lds_h no longer needed as `h`; reuse rows as per-agent vectors
#pragma unroll
  for (int t = 0; t < 2; ++t) {
    float b1v = b1[t * 16 + ln];
#pragma unroll
    for (int r = 0; r < 8; ++r) {
      int m = r + 8 * half;
      lds_h[wid][m][      t * 16 + ln] = fmaxf(aA[t][r] + b1v, 0.f);  // A1
      lds_h[wid][m][ 32 + t * 16 + ln] = aB[t][r];                    // G2 (raw)
      lds_h[wid][m][ 64 + t * 16 + ln] = aC[t][r];                    // G3 (raw)
    }
  }
  __syncthreads();

  // ------------- per-agent attention: scores, softmax, ctx ------------------
  // neigh[...,n,:] = s_n * h2  =>  n_head_n = relu(s_n*G2 + b2),
  // scores_n = <A1, n_head_n>, att = softmax_n, ctx = (sum att_n*s_n)*G3 + b3
  if (lane < 16) {
    float* v = &lds_h[wid][lane][0];
    const int g = g0 + lane;
    float sv[NN_], sc[NN_];
#pragma unroll
    for (int n = 0; n < NN_; ++n) sv[n] = sbuf[(size_t)g * NN_ + n];
#pragma unroll
    for (int n = 0; n < NN_; ++n) {
      float d = 0.f;
      for (int hh = 0; hh < H_; ++hh)
        d += v[hh] * fmaxf(sv[n] * v[32 + hh] + b2[hh], 0.f);
      sc[n] = d;
    }
    float mx = sc[0];
#pragma unroll
    for (int n = 1; n < NN_; ++n) mx = fmaxf(mx, sc[n]);
    float se = 0.f;
#pragma unroll
    for (int n = 0; n < NN_; ++n) { sc[n] = __expf(sc[n] - mx); se += sc[n]; }
    float inv = 1.f / se;
    float tt = 0.f;
#pragma unroll
    for (int n = 0; n < NN_; ++n) {
      float an = sc[n] * inv;
      att[(size_t)g * NN_ + n] = an;
      tt += an * sv[n];
    }
    for (int hh = 0; hh < H_; ++hh)
      v[96 + hh] = tt * v[64 + hh] + b3[hh];   // ctx
  }
  __syncthreads();

  // ------------- out = ctx @ W4 + b4, [16x32], K=32 -------------------------
  const float* crow = &lds_h[wid][ln][96];
  v8f ao[2];
  ao[0] = vzero8(); ao[1] = vzero8();
  for (int k0 = 0; k0 < H_; k0 += 4) {
    const int kr = k0 + 2 * half;
    v2f a; a.x = crow[kr]; a.y = crow[kr + 1];
#pragma unroll
    for (int t = 0; t < 2; ++t) {
      v2f w;
      w.x = W4[(size_t)kr * NOUT_ + t * 16 + ln];
      w.y = W4[(size_t)(kr + 1) * NOUT_ + t * 16 + ln];
      ao[t] = wmma4(a, w, ao[t]);
    }
  }
  __syncthreads();
#pragma unroll
  for (int t = 0; t < 2; ++t) {
    float bv = b4[t * 16 + ln];
#pragma unroll
    for (int r = 0; r < 8; ++r) {
      int m = r + 8 * half;
      lds_h2[wid][m][t * 16 + ln] = ao[t][r] + bv;   // out tile (reuse lds_h2)
    }
  }
  __syncthreads();

  // ------------- logits = out @ Wp + bp, [16x8], K=32 (N padded to 16) ------
  v8f al = vzero8();
  for (int k0 = 0; k0 < NOUT_; k0 += 4) {
    const int kr = k0 + 2 * half;
    v2f a; a.x = h2row[kr]; a.y = h2row[kr + 1];
    v2f w;
    w.x = (ln < NACT_) ? Wp[(size_t)kr * NACT_ + ln] : 0.f;
    w.y = (ln < NACT_) ? Wp[(size_t)(kr + 1) * NACT_ + ln] : 0.f;
    al = wmma4(a, w, al);
  }
  if (ln < NACT_) {
    float bpv = bp[ln];
#pragma unroll
    for (int r = 0; r < 8; ++r) {
      int m = r + 8 * half;
      logits[(size_t)(g0 + m) * NACT_ + ln] = al[r] + bpv;
    }
  }
}

// ---------------------------------------------------------------------------
extern "C" void kernel_launch(void* const* d_in, const int* in_sizes, int n_in,
                              void* d_out, int out_size, void* d_ws, size_t ws_size,
                              hipStream_t stream) {
  const float* x   = (const float*)d_in[0];
  const float* y   = (const float*)d_in[1];
  const float* We0 = (const float*)d_in[2];
  const float* be0 = (const float*)d_in[3];
  const float* We1 = (const float*)d_in[4];
  const float* be1 = (const float*)d_in[5];
  const float* W1  = (const float*)d_in[6];
  const float* b1  = (const float*)d_in[7];
  const float* W2  = (const float*)d_in[8];
  const float* b2  = (const float*)d_in[9];
  const float* W3  = (const float*)d_in[10];
  const float* b3  = (const float*)d_in[11];
  const float* W4  = (const float*)d_in[12];
  const float* b4  = (const float*)d_in[13];
  const float* Wp  = (const float*)d_in[14];
  const float* bp  = (const float*)d_in[15];

  float* out    = (float*)d_out;
  float* logits = out;                       // [B, A, NACT]
  float* att    = out + (size_t)B_ * A_ * NACT_;  // [B, A, NN]
  float* sbuf   = (float*)d_ws;              // [B*A*NN] row sums of y

  // 1) row-sum of y: 163840 rows of 1024, one wave each (8 waves / block)
  const int rows = B_ * A_ * NN_;
  ysum_kernel<<<rows / 8, 256, 0, stream>>>(y, sbuf);

  // 2) fused MLP + attention: 64 agents per 128-thread block
  const int blocks = (B_ * A_) / 64;
  gat_kernel<<<blocks, 128, 0, stream>>>(x, sbuf, We0, be0, We1, be1,
                                         W1, b1, W2, b2, W3, b3,
                                         W4, b4, Wp, bp, logits, att);
}